// SummarizeLayer_7834020348102
// MI455X (gfx1250) — compile-verified
//
#include <hip/hip_runtime.h>
#include <hip/hip_bf16.h>
#include <math.h>

typedef __attribute__((ext_vector_type(2))) float v2f;
typedef __attribute__((ext_vector_type(8))) float v8f;

#define DIMS 128
#define NBINS 65536
#define CAP 16384

__device__ __forceinline__ unsigned f2key(float f) {
  union { float f; unsigned u; } c; c.f = f;
  unsigned u = c.u;
  return (u & 0x80000000u) ? ~u : (u | 0x80000000u);
}
__device__ __forceinline__ float key2f(unsigned k) {
  unsigned u = (k & 0x80000000u) ? (k ^ 0x80000000u) : ~k;
  union { unsigned u; float f; } c; c.u = u;
  return c.f;
}

// K0: ||p|| -> miscF[0]
__global__ void norm_kernel(const float* __restrict__ p, float* __restrict__ miscF) {
  __shared__ float red[DIMS];
  int t = threadIdx.x;
  float v = p[t];
  red[t] = v * v;
  __syncthreads();
  for (int s = DIMS / 2; s > 0; s >>= 1) {
    if (t < s) red[t] += red[t + s];
    __syncthreads();
  }
  if (t == 0) miscF[0] = sqrtf(red[0]);
}

// K1: f32 WMMA matvec -> sortable keys + 64K-bin histogram of key>>16.
// Each wave computes 16 rows via 32x V_WMMA_F32_16X16X4_F32 with B = p-chunk
// replicated across columns (all 16 columns of D carry the same dot product).
__global__ void __launch_bounds__(256) matvec_kernel(const float* __restrict__ x,
                                                     const float* __restrict__ p,
                                                     unsigned* __restrict__ keys,
                                                     unsigned* __restrict__ hist,
                                                     int N) {
  const int lane = threadIdx.x & 31;
  const int wave = threadIdx.x >> 5;
  const int rowBase = blockIdx.x * 128 + wave * 16;
  if (rowBase >= N) return;                  // wave-uniform exit (EXEC stays full for WMMA)

  // A layout (ISA 7.12.2, 32-bit A 16x4): lanes 0-15 hold K=0,1 ; lanes 16-31 hold K=2,3
  const int m = lane & 15;
  int row = rowBase + m; if (row >= N) row = N - 1;   // clamp; result discarded later
  const int kofs = (lane >= 16) ? 2 : 0;
  const float* xr = x + (size_t)row * DIMS + kofs;
  const float* pr = p + kofs;

  v8f c = {0.f, 0.f, 0.f, 0.f, 0.f, 0.f, 0.f, 0.f};
#pragma unroll
  for (int kk = 0; kk < DIMS; kk += 4) {
    v2f a, b;
    a.x = xr[kk];     a.y = xr[kk + 1];
    b.x = pr[kk];     b.y = pr[kk + 1];     // B[k][n] = p[kk+k] for all n
    c = __builtin_amdgcn_wmma_f32_16x16x4_f32(false, a, false, b, (short)0, c, false, false);
  }

  // D layout: lane l holds D[(l>=16?8:0)+v][l&15] in c[v]; all columns equal.
  // Lanes 0-7 emit rows rowBase+0..7, lanes 16-23 emit rows rowBase+8..15.
  if ((lane & 8) == 0) {
    int v = lane & 7;
    float dotv = c[0];
#pragma unroll
    for (int q = 1; q < 8; ++q) if (v == q) dotv = c[q];
    int rowOut = rowBase + ((lane >> 4) << 3) + v;
    if (rowOut < N) {
      unsigned key = f2key(dotv);
      keys[rowOut] = key;
      atomicAdd(&hist[key >> 16], 1u);
    }
  }
}

// K2: find boundary bin B: count(bin > B) < K <= count(bin >= B)
__global__ void thresh_kernel(const unsigned* __restrict__ hist,
                              unsigned* __restrict__ misc, int K) {
  __shared__ unsigned part[256];
  __shared__ unsigned pref[256];
  int t = threadIdx.x;
  int hi = NBINS - t * 256;            // chunk t covers bins [hi-256, hi)
  unsigned s = 0;
  for (int b = hi - 1; b >= hi - 256; --b) s += hist[b];
  part[t] = s;
  __syncthreads();
  if (t == 0) {
    unsigned acc = 0;
    for (int q = 0; q < 256; ++q) { pref[q] = acc; acc += part[q]; }
  }
  __syncthreads();
  unsigned above = pref[t];
  if (above < (unsigned)K && above + part[t] >= (unsigned)K) {
    unsigned running = above;
    for (int b = hi - 1; b >= hi - 256; --b) {
      unsigned h = hist[b];
      if (running + h >= (unsigned)K) { misc[2] = (unsigned)b; misc[3] = running; break; }
      running += h;
    }
  }
}

// K3: gather candidates (bin >= B)
__global__ void compact_kernel(const unsigned* __restrict__ keys,
                               const unsigned* __restrict__ misc,
                               unsigned* __restrict__ candCount,
                               uint2* __restrict__ cand, int N) {
  int i = blockIdx.x * blockDim.x + threadIdx.x;
  if (i >= N) return;
  unsigned key = keys[i];
  if ((key >> 16) >= misc[2]) {
    unsigned pos = atomicAdd(candCount, 1u);
    if (pos < CAP) cand[pos] = make_uint2(key, (unsigned)i);
  }
}

// K4: exact stable rank (key desc, idx asc == lax.top_k tie-break); ranks<K fill output
__global__ void __launch_bounds__(256) rank_kernel(const uint2* __restrict__ cand,
                                                   const unsigned* __restrict__ misc,
                                                   unsigned* __restrict__ topIdx,
                                                   unsigned* __restrict__ topKey, int K) {
  __shared__ uint2 tile[256];
  unsigned C = misc[1]; if (C > CAP) C = CAP;
  int j = blockIdx.x * blockDim.x + threadIdx.x;
  uint2 my = make_uint2(0u, 0xffffffffu);
  if (j < (int)C) my = cand[j];
  int rank = 0;
  for (unsigned base = 0; base < C; base += 256) {
    unsigned t = base + threadIdx.x;
    tile[threadIdx.x] = (t < C) ? cand[t] : make_uint2(0u, 0xffffffffu);
    __syncthreads();
    unsigned lim = (C - base < 256u) ? (C - base) : 256u;
    if (j < (int)C) {
      for (unsigned q = 0; q < lim; ++q) {
        uint2 o = tile[q];
        rank += (o.x > my.x) || (o.x == my.x && o.y < my.y);
      }
    }
    __syncthreads();
  }
  if (j < (int)C && rank < K) { topIdx[rank] = my.y; topKey[rank] = my.x; }
}

// K5: out[j] = x[idx_j] * tanh(dot_j / ||p||)
__global__ void out_kernel(const float* __restrict__ x,
                           const unsigned* __restrict__ topIdx,
                           const unsigned* __restrict__ topKey,
                           const float* __restrict__ miscF,
                           float* __restrict__ out) {
  int j = blockIdx.x;
  int t = threadIdx.x;
  unsigned idx = topIdx[j];
  float y = key2f(topKey[j]) / miscF[0];
  float s = tanhf(y);
  out[(size_t)j * DIMS + t] = x[(size_t)idx * DIMS + t] * s;
}

extern "C" void kernel_launch(void* const* d_in, const int* in_sizes, int n_in,
                              void* d_out, int out_size, void* d_ws, size_t ws_size,
                              hipStream_t stream) {
  const float* x = (const float*)d_in[0];
  const float* p = (const float*)d_in[1];
  const int N = in_sizes[0] / DIMS;        // 500000
  const int K = out_size / DIMS;           // 2048 (k also lives on device in d_in[2])

  char* ws = (char*)d_ws;
  size_t off_hist = ((size_t)N * 4 + 255) & ~(size_t)255;   // keys at 0
  size_t off_misc = off_hist + (size_t)NBINS * 4;
  size_t off_cand = off_misc + 256;
  size_t off_tidx = off_cand + (size_t)CAP * 8;
  size_t off_tkey = off_tidx + (size_t)K * 4;

  unsigned* keys   = (unsigned*)(ws);
  unsigned* hist   = (unsigned*)(ws + off_hist);
  unsigned* misc   = (unsigned*)(ws + off_misc);
  float*    miscF  = (float*)   (ws + off_misc);
  uint2*    cand   = (uint2*)   (ws + off_cand);
  unsigned* topIdx = (unsigned*)(ws + off_tidx);
  unsigned* topKey = (unsigned*)(ws + off_tkey);

  // zero histogram + misc (counters) every call: deterministic under graph replay
  hipMemsetAsync(ws + off_hist, 0, (size_t)NBINS * 4 + 256, stream);

  norm_kernel<<<1, DIMS, 0, stream>>>(p, miscF);
  matvec_kernel<<<(N + 127) / 128, 256, 0, stream>>>(x, p, keys, hist, N);
  thresh_kernel<<<1, 256, 0, stream>>>(hist, misc, K);
  compact_kernel<<<(N + 255) / 256, 256, 0, stream>>>(keys, misc, misc + 1, cand, N);
  rank_kernel<<<CAP / 256, 256, 0, stream>>>(cand, misc, topIdx, topKey, K);
  out_kernel<<<K, DIMS, 0, stream>>>(x, topIdx, topKey, miscF, (float*)d_out);
}